// ChebNet_35716948034100
// MI455X (gfx1250) — compile-verified
//
#include <hip/hip_runtime.h>
#include <math.h>

typedef __attribute__((ext_vector_type(2))) float v2f;
typedef __attribute__((ext_vector_type(8))) float v8f;

#define TPB 256

// ---------------- elementwise helpers ----------------
__global__ void k_zero(float* __restrict__ p, long n) {
  long i = (long)blockIdx.x * blockDim.x + threadIdx.x;
  if (i < n) p[i] = 0.0f;
}
__global__ void k_copy(float* __restrict__ o, const float* __restrict__ in, long n) {
  long i = (long)blockIdx.x * blockDim.x + threadIdx.x;
  if (i < n) o[i] = in[i];
}
__global__ void k_negcopy(float* __restrict__ o, const float* __restrict__ in, long n) {
  long i = (long)blockIdx.x * blockDim.x + threadIdx.x;
  if (i < n) o[i] = -in[i];
}

// ---------------- graph normalization ----------------
__global__ void k_deg(float* __restrict__ deg, const int* __restrict__ row,
                      const int* __restrict__ col, long E) {
  long e = (long)blockIdx.x * blockDim.x + threadIdx.x;
  if (e >= E) return;
  if (row[e] != col[e]) atomicAdd(&deg[row[e]], 1.0f);
}
__global__ void k_dinv(float* __restrict__ dinv, const float* __restrict__ deg, long n) {
  long i = (long)blockIdx.x * blockDim.x + threadIdx.x;
  if (i >= n) return;
  float d = deg[i];
  dinv[i] = (d > 0.0f) ? rsqrtf(fmaxf(d, 1e-12f)) : 0.0f;
}
__global__ void k_edgew(float* __restrict__ w, const float* __restrict__ dinv,
                        const int* __restrict__ row, const int* __restrict__ col, long E) {
  long e = (long)blockIdx.x * blockDim.x + threadIdx.x;
  if (e >= E) return;
  int r = row[e], c = col[e];
  w[e] = (r != c) ? (-dinv[r] * dinv[c]) : 0.0f;
}

// pad [N,3] input into stride-4 buffer, col 3 = 0
__global__ void k_padx(float* __restrict__ o, const float* __restrict__ x, long N) {
  long t = (long)blockIdx.x * blockDim.x + threadIdx.x;
  if (t >= N * 4) return;
  long i = t >> 2;
  int c = (int)(t & 3);
  o[t] = (c < 3) ? x[i * 3 + c] : 0.0f;
}

// ---------------- SpMV: out[row,:] += scale * w[e] * in[col,:] ----------------
// C = 1<<LOGC channels; buffers have stride C. 'out' must be pre-initialized.
template <int LOGC>
__global__ void k_spmv(float* __restrict__ out, const float* __restrict__ in,
                       const int* __restrict__ row, const int* __restrict__ col,
                       const float* __restrict__ w, float scale, long E) {
  long t = (long)blockIdx.x * blockDim.x + threadIdx.x;
  long total = E << LOGC;
  if (t >= total) return;
  long e = t >> LOGC;
  int c = (int)(t & ((1 << LOGC) - 1));
  float v = in[((long)col[e] << LOGC) + c];
  atomicAdd(out + ((long)row[e] << LOGC) + c, scale * w[e] * v);
}

// ---------------- WMMA GEMM-accumulate: acc[N,32] += tx[N,S] @ Wpad[32,32] ----
// One wave computes a [16 x 32] output tile (two 16x16 N-tiles).
// A 16x4 f32 layout : lane holds A[m=lane%16, k0] / A[m, k0+1], k0 = 2*(lane/16)
// B 4x16  f32 layout: lane holds B[k0, n=lane%16] / B[k0+1, n]
// C/D 16x16 f32     : vgpr r holds D[m = r + 8*(lane/16), n = lane%16]
__global__ void k_gemm_wmma(float* __restrict__ acc, const float* __restrict__ tx,
                            const float* __restrict__ Wk, int Cin, int S, int numTiles) {
  __shared__ float Wlds[32 * 33];  // zero-padded 32x32, stride 33 vs bank conflicts
  for (int idx = threadIdx.x; idx < 32 * 32; idx += blockDim.x) {
    int r = idx >> 5, cc = idx & 31;
    Wlds[r * 33 + cc] = (r < Cin) ? Wk[r * 32 + cc] : 0.0f;
  }
  __syncthreads();

  int lane = threadIdx.x & 31;
  int gw = blockIdx.x * (blockDim.x >> 5) + (threadIdx.x >> 5);
  if (gw >= numTiles) return;  // whole-wave exit: EXEC stays all-ones for WMMA

  int m0 = gw * 16;
  int half = lane >> 4;    // 0: lanes 0-15, 1: lanes 16-31
  int mrow = lane & 15;
  int Ksteps = (Cin + 3) >> 2;

  for (int j = 0; j < 2; ++j) {  // two 16-wide N tiles -> 32 output channels
    v8f c;
#pragma unroll
    for (int r = 0; r < 8; ++r)
      c[r] = acc[(size_t)(m0 + r + 8 * half) * 32 + j * 16 + mrow];

    for (int kk = 0; kk < Ksteps; ++kk) {
      int k0 = kk * 4 + 2 * half;
      const float* ap = tx + (size_t)(m0 + mrow) * S + k0;
      v2f a = {ap[0], ap[1]};
      v2f b = {Wlds[k0 * 33 + j * 16 + mrow], Wlds[(k0 + 1) * 33 + j * 16 + mrow]};
      c = __builtin_amdgcn_wmma_f32_16x16x4_f32(false, a, false, b, (short)0, c,
                                                false, false);
    }
#pragma unroll
    for (int r = 0; r < 8; ++r)
      acc[(size_t)(m0 + r + 8 * half) * 32 + j * 16 + mrow] = c[r];
  }
}

// scalar remainder (only launched if N % 16 != 0; N=100000 divides evenly)
__global__ void k_gemm_rem(float* __restrict__ acc, const float* __restrict__ tx,
                           const float* __restrict__ Wk, int Cin, int S,
                           int startRow, int nRows) {
  int t = blockIdx.x * blockDim.x + threadIdx.x;
  if (t >= nRows * 32) return;
  int i = startRow + t / 32, o = t % 32;
  float s = 0.0f;
  for (int c = 0; c < Cin; ++c) s += tx[(size_t)i * S + c] * Wk[c * 32 + o];
  acc[(size_t)i * 32 + o] += s;
}

// h = silu(acc + b)
__global__ void k_bias_silu(float* __restrict__ h, const float* __restrict__ acc,
                            const float* __restrict__ b, long n) {
  long i = (long)blockIdx.x * blockDim.x + threadIdx.x;
  if (i >= n) return;
  float v = acc[i] + b[i & 31];
  h[i] = v / (1.0f + expf(-v));
}

// final K=1 projection [N,32] @ [32,1]
__global__ void k_outproj(float* __restrict__ out, const float* __restrict__ h,
                          const float* __restrict__ w4, int N) {
  int i = blockIdx.x * blockDim.x + threadIdx.x;
  if (i >= N) return;
  const float* hp = h + (size_t)i * 32;
  float s = 0.0f;
#pragma unroll
  for (int c = 0; c < 32; ++c) s += hp[c] * w4[c];
  out[i] = s;
}

extern "C" void kernel_launch(void* const* d_in, const int* in_sizes, int n_in,
                              void* d_out, int out_size, void* d_ws, size_t ws_size,
                              hipStream_t stream) {
  const float* x  = (const float*)d_in[0];
  const int*   ei = (const int*)d_in[1];
  // d_in[2]=batch, d_in[3]=edge_attr: unused by the reference forward
  const float* W1 = (const float*)d_in[4];
  const float* b1 = (const float*)d_in[5];
  const float* W2 = (const float*)d_in[6];
  const float* b2 = (const float*)d_in[7];
  const float* W3 = (const float*)d_in[8];
  const float* b3 = (const float*)d_in[9];
  const float* W4 = (const float*)d_in[10];

  const long N = in_sizes[0] / 3;
  const long E = in_sizes[1] / 2;
  const int K1 = in_sizes[4] / (3 * 32);
  const int K2 = in_sizes[6] / (32 * 32);
  const int K3 = in_sizes[8] / (32 * 32);
  const int* row = ei;
  const int* col = ei + E;

  // workspace carve-up (floats): 2N + E + 5*32N  (~78 MB at N=1e5, E=3.2e6)
  float* ws   = (float*)d_ws;
  float* deg  = ws;
  float* dinv = deg + N;
  float* wE   = dinv + N;
  float* B0   = wE + E;
  float* B1   = B0 + (size_t)N * 32;
  float* B2   = B1 + (size_t)N * 32;
  float* ACC  = B2 + (size_t)N * 32;
  float* H    = ACC + (size_t)N * 32;

  auto nb = [](long n) { return (unsigned)((n + TPB - 1) / TPB); };

  // ---- graph normalization (recomputed every call: deterministic) ----
  k_zero<<<nb(N), TPB, 0, stream>>>(deg, N);
  k_deg<<<nb(E), TPB, 0, stream>>>(deg, row, col, E);
  k_dinv<<<nb(N), TPB, 0, stream>>>(dinv, deg, N);
  k_edgew<<<nb(E), TPB, 0, stream>>>(wE, dinv, row, col, E);

  const long tiles = N / 16;
  const long rem = N % 16;
  auto gemm = [&](float* acc, const float* tx, const float* Wk, int Cin, int S) {
    unsigned g = (unsigned)((tiles + 7) / 8);  // 8 waves (wave32) per 256-thread block
    k_gemm_wmma<<<g, 256, 0, stream>>>(acc, tx, Wk, Cin, S, (int)tiles);
    if (rem)
      k_gemm_rem<<<nb(rem * 32), TPB, 0, stream>>>(acc, tx, Wk, Cin, S,
                                                   (int)(tiles * 16), (int)rem);
  };

  float *p0 = B0, *p1 = B1, *p2 = B2;

  // ---- layer 1: Cin=3 (padded, stride 4), K1 terms ----
  k_padx<<<nb(N * 4), TPB, 0, stream>>>(p0, x, N);
  k_zero<<<nb(N * 32), TPB, 0, stream>>>(ACC, N * 32);
  gemm(ACC, p0, W1, 3, 4);
  if (K1 > 1) {
    k_zero<<<nb(N * 4), TPB, 0, stream>>>(p1, N * 4);
    k_spmv<2><<<nb(E * 4), TPB, 0, stream>>>(p1, p0, row, col, wE, 1.0f, E);
    gemm(ACC, p1, W1 + 1 * 3 * 32, 3, 4);
    for (int k = 2; k < K1; ++k) {
      k_negcopy<<<nb(N * 4), TPB, 0, stream>>>(p2, p0, N * 4);          // Tx2 = -Tx0
      k_spmv<2><<<nb(E * 4), TPB, 0, stream>>>(p2, p1, row, col, wE, 2.0f, E);  // += 2 L Tx1
      gemm(ACC, p2, W1 + (size_t)k * 3 * 32, 3, 4);
      float* t = p0; p0 = p1; p1 = p2; p2 = t;
    }
  }
  k_bias_silu<<<nb(N * 32), TPB, 0, stream>>>(H, ACC, b1, N * 32);

  // ---- layers 2 and 3: Cin=32 (stride 32) ----
  auto layer32 = [&](const float* Wl, const float* bl, int K) {
    p0 = B0; p1 = B1; p2 = B2;
    k_copy<<<nb(N * 32), TPB, 0, stream>>>(p0, H, N * 32);
    k_zero<<<nb(N * 32), TPB, 0, stream>>>(ACC, N * 32);
    gemm(ACC, p0, Wl, 32, 32);
    if (K > 1) {
      k_zero<<<nb(N * 32), TPB, 0, stream>>>(p1, N * 32);
      k_spmv<5><<<nb(E * 32), TPB, 0, stream>>>(p1, p0, row, col, wE, 1.0f, E);
      gemm(ACC, p1, Wl + 1024, 32, 32);
      for (int k = 2; k < K; ++k) {
        k_negcopy<<<nb(N * 32), TPB, 0, stream>>>(p2, p0, N * 32);
        k_spmv<5><<<nb(E * 32), TPB, 0, stream>>>(p2, p1, row, col, wE, 2.0f, E);
        gemm(ACC, p2, Wl + (size_t)k * 1024, 32, 32);
        float* t = p0; p0 = p1; p1 = p2; p2 = t;
      }
    }
    k_bias_silu<<<nb(N * 32), TPB, 0, stream>>>(H, ACC, bl, N * 32);
  };
  layer32(W2, b2, K2);
  layer32(W3, b3, K3);

  // ---- layer 4: K=1, [32 -> 1], no bias ----
  k_outproj<<<nb(N), TPB, 0, stream>>>((float*)d_out, H, W4, (int)N);
}